// GraphAttentionLayer_74603581931798
// MI455X (gfx1250) — compile-verified
//
#include <hip/hip_runtime.h>
#include <math.h>

typedef float v2f __attribute__((ext_vector_type(2)));
typedef float v8f __attribute__((ext_vector_type(8)));

namespace {

constexpr int IN_F  = 128;
constexpr int HEADS = 4;
constexpr int DOUT  = 256;   // HEADS * 64
constexpr float NEG_SLOPE = 0.2f;
constexpr float LN_EPS = 1e-5f;

__device__ __forceinline__ unsigned encOrd(float f) {
  unsigned u = __float_as_uint(f);
  return (u & 0x80000000u) ? ~u : (u | 0x80000000u);
}
__device__ __forceinline__ float decOrd(unsigned u) {
  return (u & 0x80000000u) ? __uint_as_float(u & 0x7FFFFFFFu)
                           : __uint_as_float(~u);
}
__device__ __forceinline__ float lrelu(float x) {
  return x >= 0.f ? x : NEG_SLOPE * x;
}

// ---------------- K0: init out=bias, mx=-inf(enc), den=0 ----------------
__global__ void init_kernel(float* __restrict__ out, unsigned* __restrict__ mx,
                            float* __restrict__ den,
                            const float* __restrict__ bias, int N) {
  int i = blockIdx.x * blockDim.x + threadIdx.x;
  if (i < N * DOUT) out[i] = bias[i & (DOUT - 1)];
  if (i < N * HEADS) { mx[i] = 0x007FFFFFu; den[i] = 0.f; }
}

// ---------------- K1: h = x @ W via V_WMMA_F32_16X16X4_F32 ----------------
// 128 threads = 4 waves per block; block owns one 16-row stripe;
// wave w owns columns [64w, 64w+64) as four 16x16 accumulators.
__global__ __launch_bounds__(128) void gemm_wmma(const float* __restrict__ x,
                                                 const float* __restrict__ W,
                                                 float* __restrict__ h, int N) {
  const int wave = threadIdx.x >> 5;
  const int lane = threadIdx.x & 31;
  const int grp  = lane >> 4;     // which 16-lane half
  const int mn   = lane & 15;     // M index (A) / N index (B,C)
  const int rowBase = blockIdx.x * 16;
  int row = rowBase + mn;
  if (row >= N) row = N - 1;      // clamp loads; stores guarded
  const int colBase = wave * 64;

  v8f acc0 = {}, acc1 = {}, acc2 = {}, acc3 = {};
  const float* xr = x + (size_t)row * IN_F;
  for (int k0 = 0; k0 < IN_F; k0 += 4) {
    const int k = k0 + 2 * grp;   // A: v0=K(2g), v1=K(2g+1); lanes0-15 K0/1, lanes16-31 K2/3
    v2f a; a.x = xr[k]; a.y = xr[k + 1];
    const float* w0 = W + (size_t)k * DOUT + colBase + mn;  // B mirrors A over K
    v2f b;
    b.x = w0[0];       b.y = w0[DOUT];
    acc0 = __builtin_amdgcn_wmma_f32_16x16x4_f32(false, a, false, b, (short)0, acc0, false, false);
    b.x = w0[16];      b.y = w0[DOUT + 16];
    acc1 = __builtin_amdgcn_wmma_f32_16x16x4_f32(false, a, false, b, (short)0, acc1, false, false);
    b.x = w0[32];      b.y = w0[DOUT + 32];
    acc2 = __builtin_amdgcn_wmma_f32_16x16x4_f32(false, a, false, b, (short)0, acc2, false, false);
    b.x = w0[48];      b.y = w0[DOUT + 48];
    acc3 = __builtin_amdgcn_wmma_f32_16x16x4_f32(false, a, false, b, (short)0, acc3, false, false);
  }
  // C/D layout: VGPR r holds row M = r + 8*grp, N = lane&15
#pragma unroll
  for (int r = 0; r < 8; ++r) {
    const int m = rowBase + r + 8 * grp;
    if (m < N) {
      float* hp = h + (size_t)m * DOUT + colBase + mn;
      hp[0]  = acc0[r];
      hp[16] = acc1[r];
      hp[32] = acc2[r];
      hp[48] = acc3[r];
    }
  }
}

// ---------------- K2: a_s[n,h], a_d[n,h] = <h[n,h,:], att_{s,d}[h,:]> ------
// one wave per node; lane owns 8 consecutive d; 8-lane groups == heads.
__global__ __launch_bounds__(256) void rowdot_kernel(const float* __restrict__ h,
                                                     const float* __restrict__ att_s,
                                                     const float* __restrict__ att_d,
                                                     float* __restrict__ a_s,
                                                     float* __restrict__ a_d, int N) {
  const int w    = (blockIdx.x * blockDim.x + threadIdx.x) >> 5;
  const int lane = threadIdx.x & 31;
  if (w >= N) return;
  const float* hr = h + (size_t)w * DOUT + lane * 8;
  const float* as = att_s + lane * 8;
  const float* ad = att_d + lane * 8;
  float ps = 0.f, pd = 0.f;
#pragma unroll
  for (int j = 0; j < 8; ++j) {
    const float hv = hr[j];
    ps += hv * as[j];
    pd += hv * ad[j];
  }
#pragma unroll
  for (int m = 1; m < 8; m <<= 1) {
    ps += __shfl_xor(ps, m, 32);
    pd += __shfl_xor(pd, m, 32);
  }
  if ((lane & 7) == 0) {
    const int head = lane >> 3;
    a_s[(size_t)w * HEADS + head] = ps;
    a_d[(size_t)w * HEADS + head] = pd;
  }
}

__device__ __forceinline__ void edge_src_dst(const int* __restrict__ ei, int E,
                                             int e, int& src, int& dst) {
  if (e < E) { src = ei[e]; dst = ei[E + e]; }
  else       { src = dst = e - E; }          // self loops appended
}

// ---------------- K3: segment max of leaky-relu logits --------------------
__global__ void edge_max_kernel(const int* __restrict__ ei, int E, int N,
                                const float* __restrict__ a_s,
                                const float* __restrict__ a_d,
                                unsigned* __restrict__ mx) {
  const int e = blockIdx.x * blockDim.x + threadIdx.x;
  if (e >= E + N) return;
  int src, dst; edge_src_dst(ei, E, e, src, dst);
  const float4 as = *(const float4*)(a_s + (size_t)src * HEADS);
  const float4 ad = *(const float4*)(a_d + (size_t)dst * HEADS);
  unsigned* m = mx + (size_t)dst * HEADS;
  atomicMax(m + 0, encOrd(lrelu(as.x + ad.x)));
  atomicMax(m + 1, encOrd(lrelu(as.y + ad.y)));
  atomicMax(m + 2, encOrd(lrelu(as.z + ad.z)));
  atomicMax(m + 3, encOrd(lrelu(as.w + ad.w)));
}

// ---------------- K4: segment sum of exp(logit - mx) ----------------------
__global__ void edge_sum_kernel(const int* __restrict__ ei, int E, int N,
                                const float* __restrict__ a_s,
                                const float* __restrict__ a_d,
                                const unsigned* __restrict__ mx,
                                float* __restrict__ den) {
  const int e = blockIdx.x * blockDim.x + threadIdx.x;
  if (e >= E + N) return;
  int src, dst; edge_src_dst(ei, E, e, src, dst);
  const float4 as = *(const float4*)(a_s + (size_t)src * HEADS);
  const float4 ad = *(const float4*)(a_d + (size_t)dst * HEADS);
  const unsigned* m = mx + (size_t)dst * HEADS;
  float* dn = den + (size_t)dst * HEADS;
  atomicAdd(dn + 0, expf(lrelu(as.x + ad.x) - decOrd(m[0])));
  atomicAdd(dn + 1, expf(lrelu(as.y + ad.y) - decOrd(m[1])));
  atomicAdd(dn + 2, expf(lrelu(as.z + ad.z) - decOrd(m[2])));
  atomicAdd(dn + 3, expf(lrelu(as.w + ad.w) - decOrd(m[3])));
}

// ---------------- K5: out[dst] += alpha * h[src] (wave per edge) ----------
__global__ __launch_bounds__(256) void edge_scatter_kernel(
    const int* __restrict__ ei, int E, int N,
    const float* __restrict__ a_s, const float* __restrict__ a_d,
    const unsigned* __restrict__ mx, const float* __restrict__ den,
    const float* __restrict__ h, float* __restrict__ out) {
  const int wv   = (blockIdx.x * blockDim.x + threadIdx.x) >> 5;
  const int lane = threadIdx.x & 31;
  if (wv >= E + N) return;
  int src, dst; edge_src_dst(ei, E, wv, src, dst);
  const int head = lane >> 3;   // lane owns 8 floats -> head = lane/8
  float l = lrelu(a_s[(size_t)src * HEADS + head] +
                  a_d[(size_t)dst * HEADS + head]);
  const float alpha = expf(l - decOrd(mx[(size_t)dst * HEADS + head])) /
                      den[(size_t)dst * HEADS + head];
  const float* hp = h + (size_t)src * DOUT + lane * 8;
  float*       op = out + (size_t)dst * DOUT + lane * 8;
  const float4 h0 = *(const float4*)(hp);
  const float4 h1 = *(const float4*)(hp + 4);
  atomicAdd(op + 0, alpha * h0.x);
  atomicAdd(op + 1, alpha * h0.y);
  atomicAdd(op + 2, alpha * h0.z);
  atomicAdd(op + 3, alpha * h0.w);
  atomicAdd(op + 4, alpha * h1.x);
  atomicAdd(op + 5, alpha * h1.y);
  atomicAdd(op + 6, alpha * h1.z);
  atomicAdd(op + 7, alpha * h1.w);
}

// ---------------- K6: ELU + LayerNorm (wave per node) ---------------------
__global__ __launch_bounds__(256) void finalize_kernel(float* __restrict__ out,
                                                       const float* __restrict__ gamma,
                                                       const float* __restrict__ beta,
                                                       int N) {
  const int w    = (blockIdx.x * blockDim.x + threadIdx.x) >> 5;
  const int lane = threadIdx.x & 31;
  if (w >= N) return;
  float* op = out + (size_t)w * DOUT + lane * 8;
  float v[8];
  {
    const float4 v0 = *(const float4*)(op);
    const float4 v1 = *(const float4*)(op + 4);
    v[0] = v0.x; v[1] = v0.y; v[2] = v0.z; v[3] = v0.w;
    v[4] = v1.x; v[5] = v1.y; v[6] = v1.z; v[7] = v1.w;
  }
  float s = 0.f, sq = 0.f;
#pragma unroll
  for (int j = 0; j < 8; ++j) {
    v[j] = v[j] > 0.f ? v[j] : expm1f(v[j]);   // ELU
    s  += v[j];
    sq += v[j] * v[j];
  }
#pragma unroll
  for (int m = 1; m < 32; m <<= 1) {
    s  += __shfl_xor(s, m, 32);
    sq += __shfl_xor(sq, m, 32);
  }
  const float mean = s * (1.f / DOUT);
  const float var  = sq * (1.f / DOUT) - mean * mean;
  const float inv  = rsqrtf(var + LN_EPS);
  const int d0 = lane * 8;
#pragma unroll
  for (int j = 0; j < 8; ++j)
    v[j] = (v[j] - mean) * inv * gamma[d0 + j] + beta[d0 + j];
  *(float4*)(op)     = make_float4(v[0], v[1], v[2], v[3]);
  *(float4*)(op + 4) = make_float4(v[4], v[5], v[6], v[7]);
}

}  // namespace

extern "C" void kernel_launch(void* const* d_in, const int* in_sizes, int n_in,
                              void* d_out, int out_size, void* d_ws, size_t ws_size,
                              hipStream_t stream) {
  const float* x     = (const float*)d_in[0];
  const int*   ei    = (const int*)  d_in[1];
  const float* W     = (const float*)d_in[2];
  const float* att_s = (const float*)d_in[3];
  const float* att_d = (const float*)d_in[4];
  const float* bias  = (const float*)d_in[5];
  const float* gamma = (const float*)d_in[6];
  const float* beta  = (const float*)d_in[7];
  float* out = (float*)d_out;

  const int N = out_size / DOUT;         // 100000
  const int E = in_sizes[1] / 2;         // 1600000
  const int Etot = E + N;

  // workspace carve-up: h | a_s | a_d | den | mx
  float*    h   = (float*)d_ws;
  float*    a_s = h + (size_t)N * DOUT;
  float*    a_d = a_s + (size_t)N * HEADS;
  float*    den = a_d + (size_t)N * HEADS;
  unsigned* mx  = (unsigned*)(den + (size_t)N * HEADS);

  const int TB = 256;
  init_kernel<<<(N * DOUT + TB - 1) / TB, TB, 0, stream>>>(out, mx, den, bias, N);
  gemm_wmma<<<(N + 15) / 16, 128, 0, stream>>>(x, W, h, N);
  rowdot_kernel<<<(N * 32 + TB - 1) / TB, TB, 0, stream>>>(h, att_s, att_d, a_s, a_d, N);
  edge_max_kernel<<<(Etot + TB - 1) / TB, TB, 0, stream>>>(ei, E, N, a_s, a_d, mx);
  edge_sum_kernel<<<(Etot + TB - 1) / TB, TB, 0, stream>>>(ei, E, N, a_s, a_d, mx, den);
  edge_scatter_kernel<<<(Etot + 7) / 8, TB, 0, stream>>>(ei, E, N, a_s, a_d, mx, den, h, out);
  finalize_kernel<<<(N + 7) / 8, TB, 0, stream>>>(out, gamma, beta, N);
}